// Detached_LightXML_47296179863795
// MI455X (gfx1250) — compile-verified
//
#include <hip/hip_runtime.h>
#include <cmath>

// ---------------------------------------------------------------------------
// Detached LightXML forward, MI455X (gfx1250, wave32, WMMA f32 16x16x4)
// Outputs (flat f32, reference tuple order):
//   p0[64*8192] p1[64*256] p2[64*704] w1[64*256] w2[64*704] cands1[64*256] cands2[64*704]
// ---------------------------------------------------------------------------

typedef __attribute__((ext_vector_type(2))) float v2f;
typedef __attribute__((ext_vector_type(8))) float v8f;

#define Bsz   64
#define Hdim  512
#define G0sz  8192
#define C0sz  8
#define C1sz  11
#define K1sz  32
#define K2sz  64

__device__ __forceinline__ float sigmoidf_(float x) { return 1.0f / (1.0f + expf(-x)); }
__device__ __forceinline__ float geluf_(float x) {
  // torch nn.GELU default (erf form)
  return 0.5f * x * (1.0f + erff(x * 0.70710678118654752440f));
}

// ---- dense GEMM + bias + GELU via f32 WMMA: Hout[M,N] = gelu(A[M,K] @ W[K,N] + b) ----
__global__ void hidden_gemm_wmma(const float* __restrict__ A, const float* __restrict__ W,
                                 const float* __restrict__ bias, float* __restrict__ Hout,
                                 int K, int N) {
  const int wave = blockIdx.x * (blockDim.x >> 5) + (threadIdx.x >> 5);
  const int lane = threadIdx.x & 31;
  const int half = lane >> 4;        // 0: K=k..k+1, 1: K=k+2..k+3
  const int l16  = lane & 15;
  const int ntn  = N >> 4;
  const int m0   = (wave / ntn) << 4;
  const int n0   = (wave % ntn) << 4;
  const float* arow = A + (size_t)(m0 + l16) * K;
  v8f acc = {};
  for (int k = 0; k < K; k += 4) {
    const int ka = k + (half << 1);
    v2f a; a.x = arow[ka];                      a.y = arow[ka + 1];
    v2f b; b.x = W[(size_t)ka * N + n0 + l16];  b.y = W[(size_t)(ka + 1) * N + n0 + l16];
    acc = __builtin_amdgcn_wmma_f32_16x16x4_f32(false, a, false, b, (short)0, acc, false, false);
  }
  const int n = n0 + l16;
  const float bn = bias[n];
#pragma unroll
  for (int r = 0; r < 8; ++r) {
    const int m = m0 + r + (half << 3);
    Hout[(size_t)m * N + n] = geluf_(acc[r] + bn);
  }
}

// ---- p0 = sigmoid(h0 @ E0^T), E0 row-major [G0,H] -> B[k][n] = E0[n][k] ----
__global__ void logits0_wmma(const float* __restrict__ Hm, const float* __restrict__ E0,
                             float* __restrict__ P0) {
  const int wave = blockIdx.x * (blockDim.x >> 5) + (threadIdx.x >> 5);
  const int lane = threadIdx.x & 31;
  const int half = lane >> 4;
  const int l16  = lane & 15;
  const int m0   = (wave / (G0sz / 16)) << 4;
  const int n0   = (wave % (G0sz / 16)) << 4;
  const float* arow = Hm + (size_t)(m0 + l16) * Hdim;
  const float* erow = E0 + (size_t)(n0 + l16) * Hdim;   // contiguous K per lane
  v8f acc = {};
  for (int k = 0; k < Hdim; k += 4) {
    const int ka = k + (half << 1);
    if (k + 64 < Hdim) __builtin_prefetch(erow + ka + 64, 0, 1);  // global_prefetch_b8
    v2f a; a.x = arow[ka]; a.y = arow[ka + 1];
    v2f b; b.x = erow[ka]; b.y = erow[ka + 1];
    acc = __builtin_amdgcn_wmma_f32_16x16x4_f32(false, a, false, b, (short)0, acc, false, false);
  }
#pragma unroll
  for (int r = 0; r < 8; ++r) {
    const int m = m0 + r + (half << 3);
    P0[(size_t)m * G0sz + n0 + l16] = sigmoidf_(acc[r]);
  }
}

// ---- top-32 of p0 row + route through g0 -> cands1 ----
__global__ void topk32_route(const float* __restrict__ P0, const int* __restrict__ g0,
                             float* __restrict__ sc1, int* __restrict__ cands1_i,
                             float* __restrict__ out_cands1) {
  const int b = blockIdx.x, tid = threadIdx.x;  // 256 threads
  __shared__ float rv[256];
  __shared__ int   ri[256];
  __shared__ unsigned char taken[G0sz];
  __shared__ float ssc[K1sz];
  __shared__ int   sel[K1sz];
  const float* row = P0 + (size_t)b * G0sz;
  for (int i = tid; i < G0sz; i += 256) taken[i] = 0;
  __syncthreads();
  for (int k = 0; k < K1sz; ++k) {
    float bv = -INFINITY; int bi = 0x7fffffff;
    for (int i = tid; i < G0sz; i += 256)
      if (!taken[i]) { float v = row[i]; if (v > bv || (v == bv && i < bi)) { bv = v; bi = i; } }
    rv[tid] = bv; ri[tid] = bi; __syncthreads();
    for (int s = 128; s > 0; s >>= 1) {
      if (tid < s) {
        float v2 = rv[tid + s]; int i2 = ri[tid + s];
        if (v2 > rv[tid] || (v2 == rv[tid] && i2 < ri[tid])) { rv[tid] = v2; ri[tid] = i2; }
      }
      __syncthreads();
    }
    if (tid == 0) { ssc[k] = rv[0]; sel[k] = ri[0]; taken[ri[0]] = 1; }
    __syncthreads();
  }
  if (tid < K1sz) sc1[b * K1sz + tid] = ssc[tid];
  for (int c = tid; c < K1sz * C0sz; c += 256) {
    int cand = g0[sel[c >> 3] * C0sz + (c & 7)];
    cands1_i[b * 256 + c]  = cand;
    out_cands1[b * 256 + c] = (float)cand;
  }
}

// ---- level-1 gathered scoring: p1, w1 ----
__global__ void scored_gather_l1(const float* __restrict__ H1, const float* __restrict__ E1,
                                 const int* __restrict__ cands1_i, const float* __restrict__ sc1,
                                 float* __restrict__ out_p1, float* __restrict__ out_w1) {
  const int b = blockIdx.x, tid = threadIdx.x;  // 256 threads == 256 candidates
  __shared__ __align__(16) float hsh[Hdim];
  for (int i = tid; i < Hdim; i += 256) hsh[i] = H1[b * Hdim + i];
  __syncthreads();
  const float4* h4 = (const float4*)hsh;
  const int c = tid;
  const int cand = cands1_i[b * 256 + c];
  const float4* e4 = (const float4*)(E1 + (size_t)cand * Hdim);
  float s = 0.f;
#pragma unroll 4
  for (int q = 0; q < Hdim / 4; ++q) {
    float4 e = e4[q], h = h4[q];
    s += e.x * h.x + e.y * h.y + e.z * h.z + e.w * h.w;
  }
  const float p = sigmoidf_(s);
  out_p1[b * 256 + c] = p;
  out_w1[b * 256 + c] = p * sc1[b * K1sz + (c >> 3)];
}

// ---- top-64 of p1 row + route through g1 -> cands2 ----
__global__ void topk64_route(const float* __restrict__ P1, const int* __restrict__ cands1_i,
                             const int* __restrict__ g1, float* __restrict__ sc2,
                             int* __restrict__ cands2_i, float* __restrict__ out_cands2) {
  const int b = blockIdx.x, tid = threadIdx.x;  // 256 threads
  __shared__ float vals[256];
  __shared__ float rv[256];
  __shared__ int   ri[256];
  __shared__ float ssc[K2sz];
  __shared__ int   sel[K2sz];
  vals[tid] = P1[b * 256 + tid];
  __syncthreads();
  for (int k = 0; k < K2sz; ++k) {
    rv[tid] = vals[tid]; ri[tid] = tid; __syncthreads();
    for (int s = 128; s > 0; s >>= 1) {
      if (tid < s) {
        float v2 = rv[tid + s]; int i2 = ri[tid + s];
        if (v2 > rv[tid] || (v2 == rv[tid] && i2 < ri[tid])) { rv[tid] = v2; ri[tid] = i2; }
      }
      __syncthreads();
    }
    if (tid == 0) { ssc[k] = rv[0]; sel[k] = ri[0]; }
    __syncthreads();
    if (tid == sel[k]) vals[tid] = -INFINITY;
    __syncthreads();
  }
  if (tid < K2sz) sc2[b * K2sz + tid] = ssc[tid];
  for (int c = tid; c < K2sz * C1sz; c += 256) {
    int meta = cands1_i[b * 256 + sel[c / C1sz]];
    int cand = g1[(size_t)meta * C1sz + (c % C1sz)];
    cands2_i[b * 704 + c]   = cand;
    out_cands2[b * 704 + c] = (float)cand;
  }
}

// ---- level-2 gathered scoring: p2 (pad-row -> 0), w2 ----
__global__ void scored_gather_l2(const float* __restrict__ H2, const float* __restrict__ E2,
                                 const int* __restrict__ cands2_i, const float* __restrict__ sc2,
                                 float* __restrict__ out_p2, float* __restrict__ out_w2) {
  const int b = blockIdx.x, tid = threadIdx.x;  // 256 threads
  __shared__ __align__(16) float hsh[Hdim];
  for (int i = tid; i < Hdim; i += 256) hsh[i] = H2[b * Hdim + i];
  __syncthreads();
  const float4* h4 = (const float4*)hsh;
  for (int c = tid; c < K2sz * C1sz; c += 256) {
    const int cand = cands2_i[b * 704 + c];
    const float4* e4 = (const float4*)(E2 + (size_t)cand * Hdim);
    float s = 0.f;
    for (int q = 0; q < Hdim / 4; ++q) {
      float4 e = e4[q], h = h4[q];
      s += e.x * h.x + e.y * h.y + e.z * h.z + e.w * h.w;
    }
    const float p = (s == 0.0f) ? 0.0f : sigmoidf_(s);
    out_p2[b * 704 + c] = p;
    out_w2[b * 704 + c] = p * sc2[b * K2sz + (c / C1sz)];
  }
}

extern "C" void kernel_launch(void* const* d_in, const int* in_sizes, int n_in,
                              void* d_out, int out_size, void* d_ws, size_t ws_size,
                              hipStream_t stream) {
  const float* feat5   = (const float*)d_in[0];
  const float* feat8   = (const float*)d_in[1];
  const float* featcat = (const float*)d_in[2];
  const float* W0 = (const float*)d_in[3];
  const float* b0 = (const float*)d_in[4];
  const float* W1 = (const float*)d_in[5];
  const float* b1 = (const float*)d_in[6];
  const float* W2 = (const float*)d_in[7];
  const float* b2 = (const float*)d_in[8];
  const float* E0 = (const float*)d_in[9];
  const float* E1 = (const float*)d_in[10];
  const float* E2 = (const float*)d_in[11];
  const int*   g0 = (const int*)d_in[12];
  const int*   g1 = (const int*)d_in[13];

  float* out = (float*)d_out;
  float* p0  = out + 0;        // 524288
  float* p1  = out + 524288;   // 16384
  float* p2  = out + 540672;   // 45056
  float* w1  = out + 585728;   // 16384
  float* w2  = out + 602112;   // 45056
  float* oc1 = out + 647168;   // 16384
  float* oc2 = out + 663552;   // 45056  (total 708608)

  float* ws  = (float*)d_ws;
  float* h0  = ws;             // 64*512
  float* h1  = ws + 32768;
  float* h2  = ws + 65536;
  float* sc1 = ws + 98304;     // 64*32
  float* sc2 = ws + 100352;    // 64*64
  int*   c1i = (int*)(ws + 104448);  // 64*256
  int*   c2i = (int*)(ws + 120832);  // 64*704

  // hidden layers: 4x32 = 128 wave-tiles each, 4 waves/block
  hidden_gemm_wmma<<<32, 128, 0, stream>>>(feat5,   W0, b0, h0, 768,  Hdim);
  hidden_gemm_wmma<<<32, 128, 0, stream>>>(feat8,   W1, b1, h1, 768,  Hdim);
  hidden_gemm_wmma<<<32, 128, 0, stream>>>(featcat, W2, b2, h2, 2304, Hdim);
  // dense level-0 scoring: 4x512 = 2048 wave-tiles, 8 waves/block
  logits0_wmma<<<256, 256, 0, stream>>>(h0, E0, p0);
  topk32_route<<<Bsz, 256, 0, stream>>>(p0, g0, sc1, c1i, oc1);
  scored_gather_l1<<<Bsz, 256, 0, stream>>>(h1, E1, c1i, sc1, p1, w1);
  topk64_route<<<Bsz, 256, 0, stream>>>(p1, c1i, g1, sc2, c2i, oc2);
  scored_gather_l2<<<Bsz, 256, 0, stream>>>(h2, E2, c2i, sc2, p2, w2);
}